// NeighborAttention_28819230556412
// MI455X (gfx1250) — compile-verified
//
#include <hip/hip_runtime.h>

// ---------------------------------------------------------------------------
// NeighborAttention on gfx1250 (MI455X)
//
// Phase 1 (WMMA f32 GEMM):  Lt = latents @ W1[0:128,:] + b1   -> d_out
//                           Lb = latents @ W1[128:256,:]      -> d_ws
// Phase 2 (gather/softmax): per point n, h_k = relu(Lt[n] + Lb[nbr_k]),
//                           s_k = h_k . W2 (b2 cancels in softmax),
//                           out[n] = sum_k softmax(s)_k * latents[nbr_k]
// ---------------------------------------------------------------------------

typedef float    v2f   __attribute__((ext_vector_type(2)));
typedef float    v8f   __attribute__((ext_vector_type(8)));
typedef float    f32x4 __attribute__((ext_vector_type(4)));
typedef _Float16 f16x4 __attribute__((ext_vector_type(4)));

// ---- Lb element-type helpers (float primary path, f16 small-ws fallback) ---
__device__ __forceinline__ void storeLbElem(float* p, size_t i, float v)     { p[i] = v; }
__device__ __forceinline__ void storeLbElem(_Float16* p, size_t i, float v)  { p[i] = (_Float16)v; }

__device__ __forceinline__ f32x4 loadRow4(const float* p, size_t i4) {
  return ((const f32x4*)p)[i4];
}
__device__ __forceinline__ f32x4 loadRow4(const _Float16* p, size_t i4) {
  f16x4 h = ((const f16x4*)p)[i4];
  f32x4 r; r.x = (float)h.x; r.y = (float)h.y; r.z = (float)h.z; r.w = (float)h.w;
  return r;
}

// ---------------------------------------------------------------------------
// Phase 1: one block = 16 rows of latents, 8 waves. Wave w owns column base
// w*16 and computes BOTH the Lt tile (W1 top half) and the Lb tile (W1 bottom
// half), sharing one A fragment per k-step -> 2 WMMAs per iteration.
//
// WMMA f32 16x16x4 fragment layouts (ISA 7.12.2):
//   A (16x4): lane l(0..15)/half h: M=l, K = 2h + vgpr  -> float2 load
//   B (4x16): lane l: N=l,          K = 2h + vgpr
//   C (16x16): vgpr i: M = i + 8h, N = l
// ---------------------------------------------------------------------------
template <typename TB>
__global__ void __launch_bounds__(256)
nbr_attn_phase1(const float* __restrict__ A,    // latents [N,128]
                const float* __restrict__ W1,   // [256,128] row-major
                const float* __restrict__ b1,   // [128]
                float* __restrict__ Lt,         // [N,128] (= d_out)
                TB* __restrict__ Lb,            // [N,128] (= d_ws)
                int N)
{
  const int wave = threadIdx.x >> 5;
  const int lane = threadIdx.x & 31;
  const int half = lane >> 4;
  const int l    = lane & 15;
  const int rowBase = blockIdx.x * 16;
  const int colBase = wave * 16;            // 0..112
  if (rowBase + 16 > N) return;

  const float* arow = A  + (size_t)(rowBase + l) * 128;     // this lane's A row
  const float* btop = W1 + colBase + l;                      // W1[k][col]
  const float* bbot = W1 + (size_t)128 * 128 + colBase + l;  // W1[128+k][col]

  v8f accT = {0.f, 0.f, 0.f, 0.f, 0.f, 0.f, 0.f, 0.f};
  v8f accB = {0.f, 0.f, 0.f, 0.f, 0.f, 0.f, 0.f, 0.f};

#pragma unroll 4
  for (int kk = 0; kk < 32; ++kk) {
    const int k0 = kk * 4;
    const int kA = k0 + 2 * half;          // this lane-half's K pair start

    v2f a = *(const v2f*)(arow + kA);      // A[M=l][kA], A[M=l][kA+1]

    v2f bt, bb;
    bt.x = btop[(size_t)(kA)     * 128];
    bt.y = btop[(size_t)(kA + 1) * 128];
    bb.x = bbot[(size_t)(kA)     * 128];
    bb.y = bbot[(size_t)(kA + 1) * 128];

    accT = __builtin_amdgcn_wmma_f32_16x16x4_f32(false, a, false, bt,
                                                 (short)0, accT, false, false);
    accB = __builtin_amdgcn_wmma_f32_16x16x4_f32(false, a, false, bb,
                                                 (short)0, accB, false, false);
  }

  const int   col = colBase + l;
  const float bv  = b1[col];               // fold b1 into Lt (h = Lt + Lb)
#pragma unroll
  for (int i = 0; i < 8; ++i) {
    const int row = rowBase + i + 8 * half;
    Lt[(size_t)row * 128 + col] = accT[i] + bv;
    storeLbElem(Lb, (size_t)row * 128 + col, accB[i]);
  }
}

// ---------------------------------------------------------------------------
// Phase 2: one wave per point. Lane l owns dims 4l..4l+3 (float4), so every
// gather is one coalesced 512B row read. Scores reduced across the wave with
// xor-shuffles; softmax in registers (b2 is shift-invariant -> dropped).
// Reads its own Lt row from d_out, then overwrites it with the result.
// ---------------------------------------------------------------------------
template <typename TB>
__global__ void __launch_bounds__(256)
nbr_attn_phase2(const float* __restrict__ latents,  // [N,128]
                const int*   __restrict__ nbrs,     // [N,16]
                const float* __restrict__ W2,       // [128]
                const TB*    __restrict__ Lb,       // [N,128]
                float* out,                          // in: Lt rows, out: result
                int N)
{
  const int wave = threadIdx.x >> 5;
  const int lane = threadIdx.x & 31;
  const int n    = blockIdx.x * 8 + wave;
  if (n >= N) return;

  const f32x4 lt = loadRow4((const float*)out, (size_t)n * 32 + lane);
  const f32x4 w2 = loadRow4(W2, (size_t)lane);

  float s[16];
  int   id[16];

#pragma unroll
  for (int k = 0; k < 16; ++k) {
    const int idx = nbrs[(size_t)n * 16 + k];
    id[k] = idx;
    f32x4 lb = loadRow4(Lb, (size_t)idx * 32 + lane);
    const float hx = fmaxf(lt.x + lb.x, 0.f);
    const float hy = fmaxf(lt.y + lb.y, 0.f);
    const float hz = fmaxf(lt.z + lb.z, 0.f);
    const float hw = fmaxf(lt.w + lb.w, 0.f);
    float p = hx * w2.x + hy * w2.y + hz * w2.z + hw * w2.w;
#pragma unroll
    for (int off = 16; off; off >>= 1)      // wave32 all-reduce
      p += __shfl_xor(p, off, 32);
    s[k] = p;
  }

  float m = s[0];
#pragma unroll
  for (int k = 1; k < 16; ++k) m = fmaxf(m, s[k]);
  float sum = 0.f;
#pragma unroll
  for (int k = 0; k < 16; ++k) { s[k] = __expf(s[k] - m); sum += s[k]; }
  const float inv = 1.f / sum;

  f32x4 acc = {0.f, 0.f, 0.f, 0.f};
#pragma unroll
  for (int k = 0; k < 16; ++k) {
    const f32x4 nb = loadRow4(latents, (size_t)id[k] * 32 + lane);
    const float w  = s[k] * inv;
    acc.x += w * nb.x; acc.y += w * nb.y;
    acc.z += w * nb.z; acc.w += w * nb.w;
  }
  ((f32x4*)out)[(size_t)n * 32 + lane] = acc;
}

// ---------------------------------------------------------------------------
extern "C" void kernel_launch(void* const* d_in, const int* in_sizes, int n_in,
                              void* d_out, int out_size, void* d_ws, size_t ws_size,
                              hipStream_t stream)
{
  const float* latents   = (const float*)d_in[0];
  const int*   neighbors = (const int*)  d_in[1];
  const float* W1        = (const float*)d_in[2];
  const float* b1        = (const float*)d_in[3];
  const float* W2        = (const float*)d_in[4];
  // d_in[5] = b2: constant shift before softmax -> no effect, skipped.

  float* out = (float*)d_out;
  const int N = in_sizes[0] / 128;          // 100000

  const dim3 blk(256);
  const dim3 grid1((N + 15) / 16);
  const dim3 grid2((N + 7) / 8);

  const size_t needF32 = (size_t)N * 128 * sizeof(float);
  if (ws_size >= needF32) {
    float* Lb = (float*)d_ws;
    nbr_attn_phase1<float><<<grid1, blk, 0, stream>>>(latents, W1, b1, out, Lb, N);
    nbr_attn_phase2<float><<<grid2, blk, 0, stream>>>(latents, neighbors, W2, Lb, out, N);
  } else {
    _Float16* Lb = (_Float16*)d_ws;         // half-precision Lb if ws is tight
    nbr_attn_phase1<_Float16><<<grid1, blk, 0, stream>>>(latents, W1, b1, out, Lb, N);
    nbr_attn_phase2<_Float16><<<grid2, blk, 0, stream>>>(latents, neighbors, W2, Lb, out, N);
  }
}